// LocalEquivBordersScoreModule_11553462026623
// MI455X (gfx1250) — compile-verified
//
#include <hip/hip_runtime.h>
#include <math.h>

typedef __attribute__((ext_vector_type(2))) float v2f;
typedef __attribute__((ext_vector_type(8))) float v8f;
typedef __attribute__((ext_vector_type(4))) unsigned int u32x4;
typedef __attribute__((ext_vector_type(8))) unsigned int u32x8;

#define G_CHUNKS 8
#define J_TOTAL  2048
#define J_PER    (J_TOTAL / G_CHUNKS)
#define BDIM     128

// LDS layout (float indices): img0 | img1 | ns | t1   => 32 KB total
#define IMG0_F 0
#define IMG1_F 3072
#define NS_F   6144
#define T1_F   7168

// R[p][q] = 1 iff |p-q| <= 16  (33-wide box window, zero padded, h=w=32)
__device__ __forceinline__ float Rval(int r, int c) {
  int d = r - c;
  return (d >= -16 && d <= 16) ? 1.0f : 0.0f;
}

// Tensor Data Mover: async DMA of one image tile (3072 f32) into LDS.
// D# group0: count=1 | lds_addr | 57-bit global_addr | type=2.
// D# group1: data_size=4B, tensor_dim0=tile_dim0=3072, dim1=1, stride0=3072.
__device__ __forceinline__ void tdm_load_img(const float* gsrc, unsigned lds_byte) {
  unsigned long long ga = (unsigned long long)(size_t)gsrc;
  u32x4 g0;
  g0[0] = 1u;                                                   // count=1
  g0[1] = lds_byte;                                             // lds_addr
  g0[2] = (unsigned)ga;                                         // global_addr lo
  g0[3] = ((unsigned)(ga >> 32) & 0x01FFFFFFu) | 0x80000000u;   // ga hi | type=2
  u32x8 g1;
  g1[0] = 0x00020000u;      // workgroup_mask=0, data_size=2 (4 bytes)
  g1[1] = 3072u << 16;      // [31:16] = tensor_dim0.lo16 (=3072)
  g1[2] = 1u << 16;         // [15:0] tensor_dim0.hi16=0, [31:16] tensor_dim1.lo16=1
  g1[3] = 3072u << 16;      // [15:0] tensor_dim1.hi16=0, [31:16] tile_dim0=3072
  g1[4] = 1u;               // [15:0] tile_dim1=1, [31:16] tile_dim2=0
  g1[5] = 3072u;            // tensor_dim0_stride lo32
  g1[6] = 3072u << 16;      // stride0 hi16=0, tensor_dim1_stride lo16
  g1[7] = 0u;               // stride1 hi
  asm volatile("tensor_load_to_lds %0, %1" :: "s"(g0), "s"(g1) : "memory");
}

// One workgroup = (query b, chunk g of 256 images); 4 waves, wave32.
__global__ __launch_bounds__(BDIM) void lebs_main(
    const float* __restrict__ t_ptr,
    const float* __restrict__ x,
    const float* __restrict__ images,
    float* __restrict__ ws_m,
    float* __restrict__ ws_den,
    float* __restrict__ ws_num) {
  __shared__ float lds[8192];
  float* nsl = lds + NS_F;
  float* t1l = lds + T1_F;

  const int tid = threadIdx.x;
  const int b = blockIdx.x & 31;
  const int g = blockIdx.x >> 5;

  // cosine noise schedule (per reference)
  const float tv = t_ptr[0];
  const float co = __cosf(tv * (3.14159265358979323846f * 0.5f / 1.008f));
  const float s  = 1.0f - co * co;          // bt^2
  const float at = sqrtf(1.0f - s);         // signal scale
  const float inv2bt2 = 1.0f / (2.0f * s);

  const int lane  = tid & 31;
  const int wave  = tid >> 5;          // 4 waves: each owns one 16x16 tile
  const int lhalf = (lane >> 4) & 1;
  const int l15   = lane & 15;
  const int tI = (wave >> 1) * 16;     // output tile row base
  const int tJ = (wave & 1) * 16;      // output tile col base

  // pixels owned by this thread = stage-2 D-fragment layout
  const int col     = tJ + l15;
  const int rowbase = tI + lhalf * 8;

  // scalar-uniform "am I wave 0" (real s_cbranch: TDM ops ignore EXEC,
  // so an EXEC-masked if would issue duplicate DMAs from every wave)
  const bool wave0u = (__builtin_amdgcn_readfirstlane(wave) == 0);
  const unsigned ldsbase = (unsigned)(size_t)(void*)lds;

  // register-resident x values for the owned pixels (used by ns AND update)
  float xr0[8], xr1[8], xr2[8];
  {
    const float* xb = x + (size_t)b * 3072;
#pragma unroll
    for (int r = 0; r < 8; ++r) {
      int p = (rowbase + r) * 32 + col;
      xr0[r] = xb[p];
      xr1[r] = xb[1024 + p];
      xr2[r] = xb[2048 + p];
    }
  }

  float m[8], den[8], num0[8], num1[8], num2[8];
#pragma unroll
  for (int r = 0; r < 8; ++r) {
    m[r] = -INFINITY; den[r] = 0.f; num0[r] = 0.f; num1[r] = 0.f; num2[r] = 0.f;
  }

  const int j0 = g * J_PER;

  // prologue: DMA first image into buffer 0
  if (wave0u) {
    tdm_load_img(images + (size_t)j0 * 3072, ldsbase + IMG0_F * 4);
    __builtin_amdgcn_s_wait_tensorcnt(0);
  }
  __syncthreads();

  for (int jj = 0; jj < J_PER; ++jj) {
    const float* il = lds + ((jj & 1) ? IMG1_F : IMG0_F);

    // kick off DMA of next image into the other buffer (overlaps compute)
    if (wave0u && (jj + 1 < J_PER)) {
      tdm_load_img(images + (size_t)(j0 + jj + 1) * 3072,
                   ldsbase + ((jj & 1) ? IMG0_F : IMG1_F) * 4);
    }

    // ns[p] = sum_c (x - at*img)^2 ; keep diffs in registers for the update
    float dif0[8], dif1[8], dif2[8];
#pragma unroll
    for (int r = 0; r < 8; ++r) {
      int p = (rowbase + r) * 32 + col;
      float a0 = xr0[r] - at * il[p];
      float a1 = xr1[r] - at * il[1024 + p];
      float a2 = xr2[r] - at * il[2048 + p];
      dif0[r] = a0; dif1[r] = a1; dif2[r] = a2;
      nsl[p] = a0 * a0 + a1 * a1 + a2 * a2;
    }
    __syncthreads();

    // stage 1: T1 = ns * R   (A = ns from LDS, B = R in loop-invariant regs)
    {
      v8f acc = {0.f, 0.f, 0.f, 0.f, 0.f, 0.f, 0.f, 0.f};
#pragma unroll
      for (int kb = 0; kb < 8; ++kb) {
        int aK = kb * 4 + lhalf * 2;
        // A 16x4 layout: lane -> M = l15; v0 = A[M][aK], v1 = A[M][aK+1]
        v2f a = *(const v2f*)&nsl[(tI + l15) * 32 + aK];
        // B 4x16 layout: lane -> N = l15; v0 = B[aK][N], v1 = B[aK+1][N]
        v2f bf;
        bf.x = Rval(aK, col);
        bf.y = Rval(aK + 1, col);
        acc = __builtin_amdgcn_wmma_f32_16x16x4_f32(
            false, a, false, bf, (short)0, acc, false, false);
      }
      // D layout: VGPR r -> row = tI + lhalf*8 + r, col = tJ + l15
#pragma unroll
      for (int r = 0; r < 8; ++r)
        t1l[(tI + lhalf * 8 + r) * 32 + col] = acc[r];
    }
    __syncthreads();

    // stage 2: box = R * T1   (A = R in regs, B = T1 from LDS)
    v8f acc2 = {0.f, 0.f, 0.f, 0.f, 0.f, 0.f, 0.f, 0.f};
#pragma unroll
    for (int kb = 0; kb < 8; ++kb) {
      int aK = kb * 4 + lhalf * 2;
      v2f a;
      a.x = Rval(tI + l15, aK);
      a.y = Rval(tI + l15, aK + 1);
      v2f bf;
      bf.x = t1l[aK * 32 + col];
      bf.y = t1l[(aK + 1) * 32 + col];
      acc2 = __builtin_amdgcn_wmma_f32_16x16x4_f32(
          false, a, false, bf, (short)0, acc2, false, false);
    }

    // thread-local online softmax (per-(b,p) stabilizer cancels exactly)
#pragma unroll
    for (int r = 0; r < 8; ++r) {
      float arg = -acc2[r] * inv2bt2;
      float nm  = fmaxf(m[r], arg);
      float sc  = __expf(m[r] - nm);   // exp(-inf) = 0 on first image
      float ev  = __expf(arg - nm);
      den[r]  = den[r]  * sc + ev;
      num0[r] = num0[r] * sc + ev * dif0[r];
      num1[r] = num1[r] * sc + ev * dif1[r];
      num2[r] = num2[r] * sc + ev * dif2[r];
      m[r] = nm;
    }

    // next buffer must be resident before anyone reads it
    if (wave0u && (jj + 1 < J_PER)) __builtin_amdgcn_s_wait_tensorcnt(0);
    __syncthreads();
  }

  // write chunk partials
  const size_t base  = ((size_t)g * 32 + b) * 1024;
  const size_t base3 = ((size_t)g * 32 + b) * 3072;
#pragma unroll
  for (int r = 0; r < 8; ++r) {
    int p = (rowbase + r) * 32 + col;
    ws_m[base + p]           = m[r];
    ws_den[base + p]         = den[r];
    ws_num[base3 + p]        = num0[r];
    ws_num[base3 + 1024 + p] = num1[r];
    ws_num[base3 + 2048 + p] = num2[r];
  }
}

// Merge flash partials across chunks and finalize: out = -num/den/bt^2
__global__ void lebs_combine(const float* __restrict__ t_ptr,
                             const float* __restrict__ ws_m,
                             const float* __restrict__ ws_den,
                             const float* __restrict__ ws_num,
                             float* __restrict__ out) {
  int idx = blockIdx.x * blockDim.x + threadIdx.x;  // over 32*1024 (b,p)
  if (idx >= 32 * 1024) return;
  int b = idx >> 10;
  int p = idx & 1023;

  const float tv = t_ptr[0];
  const float co = __cosf(tv * (3.14159265358979323846f * 0.5f / 1.008f));
  const float s  = 1.0f - co * co;  // bt^2

  float M = -INFINITY;
#pragma unroll
  for (int g = 0; g < G_CHUNKS; ++g)
    M = fmaxf(M, ws_m[((size_t)g * 32 + b) * 1024 + p]);

  float den = 0.f, n0 = 0.f, n1 = 0.f, n2 = 0.f;
#pragma unroll
  for (int g = 0; g < G_CHUNKS; ++g) {
    size_t base  = ((size_t)g * 32 + b) * 1024;
    size_t base3 = ((size_t)g * 32 + b) * 3072;
    float w = __expf(ws_m[base + p] - M);
    den += w * ws_den[base + p];
    n0  += w * ws_num[base3 + p];
    n1  += w * ws_num[base3 + 1024 + p];
    n2  += w * ws_num[base3 + 2048 + p];
  }
  float inv = -1.0f / (den * s);
  out[(size_t)b * 3072 + p]        = n0 * inv;
  out[(size_t)b * 3072 + 1024 + p] = n1 * inv;
  out[(size_t)b * 3072 + 2048 + p] = n2 * inv;
}

extern "C" void kernel_launch(void* const* d_in, const int* in_sizes, int n_in,
                              void* d_out, int out_size, void* d_ws, size_t ws_size,
                              hipStream_t stream) {
  const float* t      = (const float*)d_in[0];
  const float* x      = (const float*)d_in[1];
  const float* images = (const float*)d_in[2];
  // d_in[3] is k (== 33), baked into Rval.

  float* ws     = (float*)d_ws;
  const size_t sect = (size_t)G_CHUNKS * 32 * 1024;
  float* ws_m   = ws;
  float* ws_den = ws + sect;
  float* ws_num = ws + 2 * sect;  // [G][32][3][1024]; total ws = 5 MB

  lebs_main<<<32 * G_CHUNKS, BDIM, 0, stream>>>(t, x, images, ws_m, ws_den, ws_num);
  lebs_combine<<<(32 * 1024 + 255) / 256, 256, 0, stream>>>(
      t, ws_m, ws_den, ws_num, (float*)d_out);
}